// VideoSegmentationNetwork_49460843381717
// MI455X (gfx1250) — compile-verified
//
#include <hip/hip_runtime.h>
#include <hip/hip_bf16.h>
#include <math.h>

// Problem constants (from reference setup_inputs)
#define Bb 8
#define Cc 128
#define Nn 64
#define Pp 16384   // H*W = 128*128
#define Kk 16
#define KB 128     // K-chunk staged to LDS in proto_gemm
#define EPSI 1e-8f

typedef float v2f __attribute__((ext_vector_type(2)));
typedef float v8f __attribute__((ext_vector_type(8)));

// ---------------------------------------------------------------------------
// K1: per-row softmax statistics (max, sum(exp(x-max))) over P elements
// grid = num_rows, block = 256, float4 (b128) loads
// ---------------------------------------------------------------------------
__global__ __launch_bounds__(256) void softmax_stats(const float* __restrict__ x,
                                                     float* __restrict__ stats) {
  const int row = blockIdx.x;
  const float4* xr = (const float4*)(x + (size_t)row * Pp);
  __shared__ float red[256];
  const int t = threadIdx.x;

  float mx = -__builtin_inff();
  for (int i = t; i < Pp / 4; i += 256) {
    const float4 v = xr[i];
    mx = fmaxf(mx, fmaxf(fmaxf(v.x, v.y), fmaxf(v.z, v.w)));
  }
  red[t] = mx; __syncthreads();
  for (int s = 128; s > 0; s >>= 1) {
    if (t < s) red[t] = fmaxf(red[t], red[t + s]);
    __syncthreads();
  }
  mx = red[0]; __syncthreads();

  float sm = 0.f;
  for (int i = t; i < Pp / 4; i += 256) {
    const float4 v = xr[i];
    sm += expf(v.x - mx) + expf(v.y - mx) + expf(v.z - mx) + expf(v.w - mx);
  }
  red[t] = sm; __syncthreads();
  for (int s = 128; s > 0; s >>= 1) {
    if (t < s) red[t] += red[t + s];
    __syncthreads();
  }
  if (t == 0) { stats[row * 2] = mx; stats[row * 2 + 1] = red[0]; }
}

// ---------------------------------------------------------------------------
// K2: prototype GEMM with fused softmax on A operand, deterministic K-split.
//   partial[b][ks][m][c] = sum_{p in split ks} softmax(A[b][m][:])[p] * F[b][c][p]
// A: [B][M][P], F: [B][C][P], stats: [B*M][2]
// block = 256 (8 waves, 2x4 tiles of 16x16 -> 32(M) x 64(C) per WG)
// grid = (C/64, M/32, B*KS)
// ---------------------------------------------------------------------------
__global__ __launch_bounds__(256) void proto_gemm(const float* __restrict__ A,
                                                  const float* __restrict__ F,
                                                  const float* __restrict__ stats,
                                                  float* __restrict__ out,
                                                  int M, int KS) {
  const int b  = blockIdx.z / KS;
  const int ks = blockIdx.z % KS;
  const int m0 = blockIdx.y * 32;
  const int c0 = blockIdx.x * 64;
  const int pbeg = ks * (Pp / KS);
  const int pend = pbeg + (Pp / KS);

  __shared__ float As[32][KB + 4];   // stride 132: b128-aligned + conflict-free cols
  __shared__ float Fs[64][KB + 4];

  const int tid  = threadIdx.x;
  const int lane = tid & 31;
  const int wave = tid >> 5;
  const int wm = (wave >> 2) * 16;  // 0 / 16
  const int wc = (wave & 3) * 16;   // 0..48

  const float* Ab = A + ((size_t)b * M + m0) * Pp;
  const float* Fb = F + ((size_t)b * Cc + c0) * Pp;
  const float* st = stats + ((size_t)b * M + m0) * 2;

  v8f acc = {};
  const int half = (lane < 16) ? 0 : 2;
  const int l16  = lane & 15;

  for (int p0 = pbeg; p0 < pend; p0 += KB) {
    // stage A chunk (32 x KB) with softmax applied, b128 loads
    for (int i = tid; i < 32 * (KB / 4); i += 256) {
      const int r  = i >> 5;            // KB/4 = 32 quads per row
      const int c4 = (i & 31) << 2;
      const float4 v = *(const float4*)(Ab + (size_t)r * Pp + p0 + c4);
      const float mx = st[r * 2];
      const float rs = 1.0f / st[r * 2 + 1];
      float4 o;
      o.x = expf(v.x - mx) * rs; o.y = expf(v.y - mx) * rs;
      o.z = expf(v.z - mx) * rs; o.w = expf(v.w - mx) * rs;
      *(float4*)&As[r][c4] = o;
    }
    // stage F chunk (64 x KB), b128 loads
    for (int i = tid; i < 64 * (KB / 4); i += 256) {
      const int r  = i >> 5;
      const int c4 = (i & 31) << 2;
      *(float4*)&Fs[r][c4] = *(const float4*)(Fb + (size_t)r * Pp + p0 + c4);
    }
    __syncthreads();

#pragma unroll
    for (int kk = 0; kk < KB; kk += 4) {
      v2f a, bf;
      a.x  = As[wm + l16][kk + half];
      a.y  = As[wm + l16][kk + half + 1];
      bf.x = Fs[wc + l16][kk + half];
      bf.y = Fs[wc + l16][kk + half + 1];
      acc = __builtin_amdgcn_wmma_f32_16x16x4_f32(false, a, false, bf,
                                                  (short)0, acc, false, false);
    }
    __syncthreads();
  }

  float* ob = out + (((size_t)b * KS + ks) * M + m0 + wm) * Cc + c0 + wc;
  const int rbase = (lane < 16) ? 0 : 8;
#pragma unroll
  for (int i = 0; i < 8; i++)
    ob[(size_t)(rbase + i) * Cc + l16] = acc[i];
}

// ---------------------------------------------------------------------------
// K2b: fixed-order K-split reduction: out[b][m][c] = sum_ks partial[b][ks][m][c]
// grid = B*MC/256, block = 256
// ---------------------------------------------------------------------------
__global__ __launch_bounds__(256) void ksplit_reduce(const float* __restrict__ part,
                                                     float* __restrict__ out,
                                                     int MC, int KS) {
  const int i = blockIdx.x * 256 + threadIdx.x;   // over B*MC
  const int b = i / MC, rem = i % MC;
  const float* p = part + (size_t)b * KS * MC + rem;
  float s = 0.f;
  for (int k = 0; k < KS; k++) s += p[(size_t)k * MC];
  out[i] = s;
}

// ---------------------------------------------------------------------------
// K3a: inverse L2 norm of rows of length 128. grid = rows, block = 128
// ---------------------------------------------------------------------------
__global__ __launch_bounds__(128) void row_invnorm(const float* __restrict__ x,
                                                   float* __restrict__ inv) {
  const int row = blockIdx.x;
  __shared__ float red[128];
  const int t = threadIdx.x;
  const float v = x[(size_t)row * Cc + t];
  red[t] = v * v; __syncthreads();
  for (int s = 64; s > 0; s >>= 1) {
    if (t < s) red[t] += red[t + s];
    __syncthreads();
  }
  if (t == 0) inv[row] = 1.0f / (sqrtf(red[0]) + EPSI);
}

// ---------------------------------------------------------------------------
// K3b: per-pixel inverse channel norm of features_t: inv[b][p], float4 per thread
// grid = (P/1024, B), block = 256
// ---------------------------------------------------------------------------
__global__ __launch_bounds__(256) void col_invnorm(const float* __restrict__ feat,
                                                   float* __restrict__ inv) {
  const int b  = blockIdx.y;
  const int p4 = blockIdx.x * 256 + threadIdx.x;           // float4 index
  const float4* fb = (const float4*)(feat + (size_t)b * Cc * Pp) + p4;
  float4 s = make_float4(0.f, 0.f, 0.f, 0.f);
  for (int c = 0; c < Cc; c++) {
    const float4 v = fb[(size_t)c * (Pp / 4)];
    s.x += v.x * v.x; s.y += v.y * v.y; s.z += v.z * v.z; s.w += v.w * v.w;
  }
  float4 o;
  o.x = 1.0f / (sqrtf(s.x) + EPSI); o.y = 1.0f / (sqrtf(s.y) + EPSI);
  o.z = 1.0f / (sqrtf(s.z) + EPSI); o.w = 1.0f / (sqrtf(s.w) + EPSI);
  ((float4*)inv)[(size_t)b * (Pp / 4) + p4] = o;
}

// ---------------------------------------------------------------------------
// K4: cosine sim + top-16 (descending, low-index tiebreak) + gather
// filtered[b][j][c][n] = ref_protos[b][idx(b,n,j)][c]
// grid = (N, B), block = 128
// ---------------------------------------------------------------------------
__global__ __launch_bounds__(128) void sim_topk_gather(const float* __restrict__ tp,
                                                       const float* __restrict__ rp,
                                                       const float* __restrict__ invt,
                                                       const float* __restrict__ invr,
                                                       float* __restrict__ filtered) {
  const int b = blockIdx.y, n = blockIdx.x, t = threadIdx.x;
  __shared__ float tn[Cc];
  __shared__ float sim[Cc];
  __shared__ float rv[Cc];
  __shared__ int   ri[Cc];
  __shared__ int   chosen[Kk];

  tn[t] = tp[((size_t)b * Nn + n) * Cc + t] * invt[b * Nn + n];
  __syncthreads();

  const float* rpb = rp + (size_t)b * Cc * Cc;
  float s = 0.f;
  for (int c = 0; c < Cc; c++) s += tn[c] * rpb[(size_t)t * Cc + c];
  sim[t] = s * invr[b * Cc + t];
  __syncthreads();

  for (int j = 0; j < Kk; j++) {
    rv[t] = sim[t]; ri[t] = t; __syncthreads();
    for (int sft = 64; sft > 0; sft >>= 1) {
      if (t < sft) {
        const float v2 = rv[t + sft]; const int i2 = ri[t + sft];
        if (v2 > rv[t] || (v2 == rv[t] && i2 < ri[t])) { rv[t] = v2; ri[t] = i2; }
      }
      __syncthreads();
    }
    if (t == 0) { chosen[j] = ri[0]; sim[ri[0]] = -__builtin_inff(); }
    __syncthreads();
  }

  for (int j = 0; j < Kk; j++) {
    const int idx = chosen[j];
    filtered[(((size_t)b * Kk + j) * Cc + t) * Nn + n] = rpb[(size_t)idx * Cc + t];
  }
}

// ---------------------------------------------------------------------------
// K5: corr[b][n][p] = sum_c (tp[b][n][c]*invt[b][n]) * (feat[b][c][p]*invf[b][p])
// block = 256 (8 waves, 4(M) x 2(P) tiles -> 64 x 32 per WG), grid = (P/32, B)
// ---------------------------------------------------------------------------
__global__ __launch_bounds__(256) void corr_gemm(const float* __restrict__ tp,
                                                 const float* __restrict__ feat,
                                                 const float* __restrict__ invt,
                                                 const float* __restrict__ invf,
                                                 float* __restrict__ corr) {
  const int b  = blockIdx.y;
  const int p0 = blockIdx.x * 32;

  __shared__ float As[Nn][Cc + 4];  // [n][c], stride 132
  __shared__ float Bs[Cc][36];      // [c][p], stride 36

  const int tid  = threadIdx.x;
  const int lane = tid & 31;
  const int wave = tid >> 5;
  const int wm = (wave & 3) * 16;   // 0..48 (covers all 64 rows)
  const int wp = (wave >> 2) * 16;  // 0 / 16

  // stage A (64x128) scaled by row inv-norm, b128 loads
  const float* tpb = tp + (size_t)b * Nn * Cc;
  const float* itb = invt + (size_t)b * Nn;
  for (int i = tid; i < Nn * (Cc / 4); i += 256) {   // 2048 quads
    const int r  = i >> 5;            // Cc/4 = 32 quads per row
    const int c4 = (i & 31) << 2;
    float4 v = *(const float4*)(tpb + (size_t)r * Cc + c4);
    const float sc = itb[r];
    v.x *= sc; v.y *= sc; v.z *= sc; v.w *= sc;
    *(float4*)&As[r][c4] = v;
  }
  // stage B (128x32) scaled by per-pixel inv-norm, b128 loads
  const float* fb  = feat + (size_t)b * Cc * Pp;
  const float* ifb = invf + (size_t)b * Pp + p0;
  for (int i = tid; i < Cc * 8; i += 256) {          // 1024 quads
    const int r  = i >> 3;            // 8 quads per row
    const int c4 = (i & 7) << 2;
    float4 v = *(const float4*)(fb + (size_t)r * Pp + p0 + c4);
    v.x *= ifb[c4 + 0]; v.y *= ifb[c4 + 1];
    v.z *= ifb[c4 + 2]; v.w *= ifb[c4 + 3];
    *(float4*)&Bs[r][c4] = v;
  }
  __syncthreads();

  const int half = (lane < 16) ? 0 : 2;
  const int l16  = lane & 15;
  v8f acc = {};
#pragma unroll
  for (int kk = 0; kk < Cc; kk += 4) {
    v2f a, bf;
    a.x  = As[wm + l16][kk + half];
    a.y  = As[wm + l16][kk + half + 1];
    bf.x = Bs[kk + half][wp + l16];
    bf.y = Bs[kk + half + 1][wp + l16];
    acc = __builtin_amdgcn_wmma_f32_16x16x4_f32(false, a, false, bf,
                                                (short)0, acc, false, false);
  }

  float* ob = corr + ((size_t)b * Nn + wm) * Pp + p0 + wp;
  const int rbase = (lane < 16) ? 0 : 8;
#pragma unroll
  for (int i = 0; i < 8; i++)
    ob[(size_t)(rbase + i) * Pp + l16] = acc[i];
}

// ---------------------------------------------------------------------------
extern "C" void kernel_launch(void* const* d_in, const int* in_sizes, int n_in,
                              void* d_out, int out_size, void* d_ws, size_t ws_size,
                              hipStream_t stream) {
  const float* feat_t  = (const float*)d_in[0];
  const float* feat_r  = (const float*)d_in[1];
  const float* masks_t = (const float*)d_in[2];
  // d_in[3] is k == 16, fixed at compile time (Kk)

  float* ws      = (float*)d_ws;
  float* stats_t = ws;                               // B*N*2   = 1024
  float* stats_r = stats_t + Bb * Nn * 2;            // B*C*2   = 2048
  float* tp      = stats_r + Bb * Cc * 2;            // B*N*C   = 65536
  float* rp      = tp + (size_t)Bb * Nn * Cc;        // B*C*C   = 131072
  float* invt    = rp + (size_t)Bb * Cc * Cc;        // B*N     = 512
  float* invr    = invt + Bb * Nn;                   // B*C     = 1024
  float* invf    = invr + Bb * Cc;                   // B*P     = 131072
  float* base_end = invf + (size_t)Bb * Pp;
  const size_t base_floats = (size_t)(base_end - ws);

  // deterministic K-split factor (fixed-order reduce; no float atomics)
  int KS = 8;
  const size_t split_floats = (size_t)Bb * KS * Nn * Cc + (size_t)Bb * KS * Cc * Cc;
  if (ws_size < (base_floats + split_floats) * sizeof(float)) KS = 1;

  float* part_t = (KS > 1) ? base_end : tp;
  float* part_r = (KS > 1) ? base_end + (size_t)Bb * KS * Nn * Cc : rp;

  float* out      = (float*)d_out;
  float* filtered = out;                              // [B,k,C,N]
  float* corr     = out + (size_t)Bb * Kk * Cc * Nn;  // [B,N,H,W]

  // 1) softmax statistics for masks_t rows and features_r rows
  softmax_stats<<<Bb * Nn, 256, 0, stream>>>(masks_t, stats_t);
  softmax_stats<<<Bb * Cc, 256, 0, stream>>>(feat_r, stats_r);

  // 2) prototype GEMMs (fused softmax on A), K-split for occupancy
  proto_gemm<<<dim3(Cc / 64, Nn / 32, Bb * KS), 256, 0, stream>>>(
      masks_t, feat_t, stats_t, part_t, Nn, KS);
  proto_gemm<<<dim3(Cc / 64, Cc / 32, Bb * KS), 256, 0, stream>>>(
      feat_r, feat_r, stats_r, part_r, Cc, KS);
  if (KS > 1) {
    ksplit_reduce<<<(Bb * Nn * Cc) / 256, 256, 0, stream>>>(part_t, tp, Nn * Cc, KS);
    ksplit_reduce<<<(Bb * Cc * Cc) / 256, 256, 0, stream>>>(part_r, rp, Cc * Cc, KS);
  }

  // 3) norms
  row_invnorm<<<Bb * Nn, 128, 0, stream>>>(tp, invt);
  row_invnorm<<<Bb * Cc, 128, 0, stream>>>(rp, invr);
  col_invnorm<<<dim3(Pp / 1024, Bb), 256, 0, stream>>>(feat_t, invf);

  // 4) cosine KNN + gather -> filtered [B,k,C,N]
  sim_topk_gather<<<dim3(Nn, Bb), 128, 0, stream>>>(tp, rp, invt, invr, filtered);

  // 5) normalized correlation map -> corr [B,N,H,W]
  corr_gemm<<<dim3(Pp / 32, Bb), 256, 0, stream>>>(tp, feat_t, invt, invf, corr);
}